// NoisyTopkRouter_86835648791007
// MI455X (gfx1250) — compile-verified
//
#include <hip/hip_runtime.h>
#include <hip/hip_bf16.h>
#include <math.h>

// NoisyTopkRouter, fused single kernel for MI455X (gfx1250, wave32).
//
// Problem: B=4, S=8192, D=2048, E=8, k=2.
//   gate  = x @ Wg^T + bg            [B,S,E]
//   noisy = noise*softplus(x@Wn^T+bn) + gate
//   top-2 over E, softmax over the 2, scatter back; also emit indices.
//
// Design: both projections fused into ONE GEMM with a combined [D x 16]
// B-matrix (cols 0..7 = Wg experts, cols 8..15 = Wn experts), computed with
// V_WMMA_F32_16X16X4_F32 (full fp32, matches reference precision).
// Bandwidth-bound: x (256 MB) read exactly once -> ~11 us floor @ 23.3 TB/s.
// x / noise are streamed with NT temporal hints (single-use, larger than L2);
// the 128 KB of weights stay regular-temporal so they remain cache-resident
// for all 2048 tile-waves.

typedef float v2f __attribute__((ext_vector_type(2)));
typedef float v8f __attribute__((ext_vector_type(8)));

#define B_  4
#define S_  8192
#define D_  2048
#define E_  8
#define K_  2
#define T_  (B_ * S_)              // 32768 tokens
#define WAVES_PER_BLOCK 8
#define TOKENS_PER_WAVE 16

__global__ __launch_bounds__(256)
void noisy_topk_router_kernel(const float* __restrict__ x,
                              const float* __restrict__ Wg,
                              const float* __restrict__ bg,
                              const float* __restrict__ Wn,
                              const float* __restrict__ bn,
                              const float* __restrict__ noise,
                              float* __restrict__ router_out,   // [T, E]
                              float* __restrict__ experts_out)  // [T, k] (as float)
{
    // Per-wave 16x16 fp32 tile scratch; inner pad 18 keeps both LDS stores
    // (halves offset by 8 rows) and the per-token reads bank-conflict-free.
    __shared__ float lds_c[WAVES_PER_BLOCK][TOKENS_PER_WAVE][18];

    const int tid   = threadIdx.x;
    const int wave  = tid >> 5;
    const int lane  = tid & 31;
    const int m     = lane & 15;    // A: row M; B: column N; C: column N
    const int khalf = lane >> 4;    // 0 or 1
    const int koff  = khalf * 2;    // K sub-offset per ISA 32-bit A/B layout

    const int tile   = blockIdx.x * WAVES_PER_BLOCK + wave;
    const int token0 = tile * TOKENS_PER_WAVE;

    // A fragment source: x[token0+m][k + koff .. +1]  (contiguous float2)
    const float* aptr = x + (size_t)(token0 + m) * D_ + koff;
    // B fragment source: combined weight column m: Wg rows for m<8, Wn for m>=8.
    const float* wrow = (m < E_) ? (Wg + (size_t)m * D_)
                                 : (Wn + (size_t)(m - E_) * D_);
    const float* bptr = wrow + koff;

    v8f c = {0.f, 0.f, 0.f, 0.f, 0.f, 0.f, 0.f, 0.f};

    // K loop: 512 x v_wmma_f32_16x16x4_f32 accumulating both projections.
#pragma unroll 8
    for (int k = 0; k < D_; k += 4) {
        // x is single-use streaming data (256 MB > L2): non-temporal load.
        v2f a = __builtin_nontemporal_load((const v2f*)(aptr + k));
        // weights are hot (reused by every wave): regular-temporal load.
        v2f b = *(const v2f*)(bptr + k);
        // (neg_a, A, neg_b, B, c_mod, C, reuse_a, reuse_b)
        c = __builtin_amdgcn_wmma_f32_16x16x4_f32(
                false, a, false, b, (short)0, c, false, false);
    }

    // C layout: VGPR j -> (M = j + 8*khalf, N = m). Spill tile to LDS.
#pragma unroll
    for (int j = 0; j < 8; ++j)
        lds_c[wave][j + 8 * khalf][m] = c[j];

    __syncthreads();

    // Epilogue: 16 lanes per wave, one token each.
    if (lane < TOKENS_PER_WAVE) {
        const int token = token0 + lane;

        float v1 = -INFINITY, v2 = -INFINITY;
        int   i1 = 0, i2 = 0;
#pragma unroll
        for (int e = 0; e < E_; ++e) {
            float g    = lds_c[wave][lane][e] + bg[e];
            float npre = lds_c[wave][lane][E_ + e] + bn[e];
            // numerically stable softplus = max(x,0) + log1p(exp(-|x|))
            float sp = fmaxf(npre, 0.f) + log1pf(expf(-fabsf(npre)));
            // noise is also single-use streaming data: NT load.
            float nz = __builtin_nontemporal_load(noise + (size_t)token * E_ + e);
            float nl = nz * sp + g;
            // stable top-2: strict '>' keeps the lower index on ties,
            // matching jax.lax.top_k / torch.topk ordering.
            if (nl > v1)      { v2 = v1; i2 = i1; v1 = nl; i1 = e; }
            else if (nl > v2) { v2 = nl; i2 = e; }
        }

        // softmax over the two kept logits
        float e2 = expf(v2 - v1);
        float inv = 1.f / (1.f + e2);
        float p1 = inv;
        float p2 = e2 * inv;

        // Outputs are write-once, never re-read by this grid: NT stores.
        float* ro = router_out + (size_t)token * E_;
#pragma unroll
        for (int e = 0; e < E_; ++e) {
            float v = (e == i1) ? p1 : ((e == i2) ? p2 : 0.f);
            __builtin_nontemporal_store(v, ro + e);
        }

        // indices output (int32 in reference; d_out is a float32 buffer for a
        // tuple return, small ints are exactly representable)
        __builtin_nontemporal_store((float)i1, experts_out + (size_t)token * K_ + 0);
        __builtin_nontemporal_store((float)i2, experts_out + (size_t)token * K_ + 1);
    }
}

extern "C" void kernel_launch(void* const* d_in, const int* in_sizes, int n_in,
                              void* d_out, int out_size, void* d_ws, size_t ws_size,
                              hipStream_t stream)
{
    (void)in_sizes; (void)n_in; (void)out_size; (void)d_ws; (void)ws_size;

    const float* x     = (const float*)d_in[0];
    const float* Wg    = (const float*)d_in[1];
    const float* bg    = (const float*)d_in[2];
    const float* Wn    = (const float*)d_in[3];
    const float* bn    = (const float*)d_in[4];
    const float* noise = (const float*)d_in[5];
    // d_in[6] is k (scalar, == 2), baked into the kernel.

    float* router  = (float*)d_out;                       // T_*E_ floats
    float* experts = (float*)d_out + (size_t)T_ * E_;     // T_*K_ entries

    const int blocks = T_ / (WAVES_PER_BLOCK * TOKENS_PER_WAVE);  // 256
    noisy_topk_router_kernel<<<blocks, 256, 0, stream>>>(
        x, Wg, bg, Wn, bn, noise, router, experts);
}